// NucleusIsthmi_88149908783720
// MI455X (gfx1250) — compile-verified
//
#include <hip/hip_runtime.h>

// Izhikevich 'IB' dual-population simulation, MI455X (gfx1250).
// Memory-bound streaming kernel: ~576 MiB traffic -> ~25us floor @ 23.3 TB/s.
// Noise rows are staged via CDNA5 async global->LDS B128 copies (ASYNCcnt)
// with double buffering; all neuron state lives in VGPRs for all 15 steps.

#define STEPS 15
#define TPB   256      // threads per block (8 wave32 waves)
#define ELPT  4        // elements (floats) per thread, via float4

__device__ __forceinline__ void izh_step(float& v, float& u, float& r, float I) {
    // v_new = v + (0.04 v^2 + 5 v + 140 - u + I), dt = 1
    float vn = v + (v * v * 0.04f + 5.0f * v + 140.0f - u + I);
    // u_new = u + a*(b*v - u), a=0.02, b=0.2 (uses pre-update v)
    float un = u + 0.02f * (0.2f * v - u);
    bool spk = vn >= 30.0f;
    v = spk ? -55.0f : vn;              // c = -55
    u = spk ? (un + 4.0f) : un;         // d = 4
    r = r + 0.05f * ((spk ? 1.0f : 0.0f) - r);   // EMA rate, alpha=0.05
}

__global__ __launch_bounds__(TPB) void izh_kernel(
    const float* __restrict__ sgcL, const float* __restrict__ sgcR,
    const float* __restrict__ noise,
    const float* __restrict__ v0, const float* __restrict__ u0,
    const float* __restrict__ rate0,
    float* __restrict__ partL, float* __restrict__ partR, int n)
{
    __shared__ float4 nbuf[2][TPB];     // double-buffered noise tiles (8 KiB)
    __shared__ float  redL[TPB];
    __shared__ float  redR[TPB];

    const int tid = threadIdx.x;
    const int gid = blockIdx.x * TPB + tid;   // float4 index
    // scalar drive currents (uniform; compiler lowers to s_load)
    const float IL = sgcL[0] * 60.0f - 3.0f;  // INPUT_GAIN=60, I_TONIC=-3
    const float IR = sgcR[0] * 60.0f - 3.0f;

    // neuron state in registers for the whole simulation
    float4 vL = ((const float4*)v0)[gid];
    float4 uL = ((const float4*)u0)[gid];
    float4 rL = ((const float4*)rate0)[gid];
    float4 vR = vL, uR = uL, rR = rL;

    // LDS byte offsets (workgroup-relative, addr[31:0]) of this lane's slots
    const unsigned lds0 = (unsigned)(size_t)&nbuf[0][tid];
    const unsigned lds1 = (unsigned)(size_t)&nbuf[1][tid];

    const int stride_b = n * 4;               // bytes between noise rows
    int goff = gid * 16;                      // byte offset of step-0 tile

    // prologue: async-prefetch step 0 noise into buffer 0 (ASYNCcnt++)
    asm volatile("global_load_async_to_lds_b128 %0, %1, %2"
                 :: "v"(lds0), "v"(goff), "s"(noise) : "memory");

    for (int s = 0; s < STEPS; ++s) {
        const int cur = s & 1;
        if (s + 1 < STEPS) {
            const unsigned lds_nxt = cur ? lds0 : lds1;
            const int goff_nxt = goff + stride_b;
            // make sure last iteration's LDS reads of the buffer we are about
            // to overwrite have completed
            asm volatile("s_wait_dscnt 0x0" ::: "memory");
            asm volatile("global_load_async_to_lds_b128 %0, %1, %2"
                         :: "v"(lds_nxt), "v"(goff_nxt), "s"(noise) : "memory");
            // async loads complete in order: <=1 outstanding => step-s tile ready
            asm volatile("s_wait_asynccnt 0x1" ::: "memory");
        } else {
            asm volatile("s_wait_asynccnt 0x0" ::: "memory");
        }

        float4 nz4 = nbuf[cur][tid];
        goff += stride_b;

        // NOISE_STD = 0.5; same sample feeds both populations
#define IZH_LANE(comp)                                             \
        {                                                          \
            float nz = nz4.comp * 0.5f;                            \
            izh_step(vL.comp, uL.comp, rL.comp, IL + nz);          \
            izh_step(vR.comp, uR.comp, rR.comp, IR + nz);          \
        }
        IZH_LANE(x) IZH_LANE(y) IZH_LANE(z) IZH_LANE(w)
#undef IZH_LANE
    }

    // deterministic block reduction of final rates
    redL[tid] = rL.x + rL.y + rL.z + rL.w;
    redR[tid] = rR.x + rR.y + rR.z + rR.w;
    __syncthreads();
    for (int off = TPB / 2; off > 0; off >>= 1) {
        if (tid < off) {
            redL[tid] += redL[tid + off];
            redR[tid] += redR[tid + off];
        }
        __syncthreads();
    }
    if (tid == 0) {
        partL[blockIdx.x] = redL[0];
        partR[blockIdx.x] = redR[0];
    }
}

__global__ __launch_bounds__(1024) void reduce_kernel(
    const float* __restrict__ partL, const float* __restrict__ partR,
    int nblocks, float inv_n, float* __restrict__ out)
{
    __shared__ float sL[1024];
    __shared__ float sR[1024];
    const int tid = threadIdx.x;
    float aL = 0.0f, aR = 0.0f;
    for (int i = tid; i < nblocks; i += 1024) {   // fixed order -> deterministic
        aL += partL[i];
        aR += partR[i];
    }
    sL[tid] = aL; sR[tid] = aR;
    __syncthreads();
    for (int off = 512; off > 0; off >>= 1) {
        if (tid < off) {
            sL[tid] += sL[tid + off];
            sR[tid] += sR[tid + off];
        }
        __syncthreads();
    }
    if (tid == 0) {
        float mL = sL[0] * inv_n;
        float mR = sR[0] * inv_n;
        out[0] = mL;
        out[1] = mR;
        out[2] = -mR * 3.0f;   // FB_GAIN = 3
        out[3] = -mL * 3.0f;
    }
}

extern "C" void kernel_launch(void* const* d_in, const int* in_sizes, int n_in,
                              void* d_out, int out_size, void* d_ws, size_t ws_size,
                              hipStream_t stream)
{
    const float* sgcL  = (const float*)d_in[0];
    const float* sgcR  = (const float*)d_in[1];
    const float* noise = (const float*)d_in[2];
    const float* v0    = (const float*)d_in[3];
    const float* u0    = (const float*)d_in[4];
    const float* rate0 = (const float*)d_in[5];

    const int n = in_sizes[3];                 // 8,388,608
    const int nblocks = n / (TPB * ELPT);      // 8192

    float* partL = (float*)d_ws;               // fully rewritten every call
    float* partR = partL + nblocks;

    izh_kernel<<<nblocks, TPB, 0, stream>>>(sgcL, sgcR, noise,
                                            v0, u0, rate0, partL, partR, n);
    reduce_kernel<<<1, 1024, 0, stream>>>(partL, partR, nblocks,
                                          1.0f / (float)n, (float*)d_out);
}